// AttentionRouting_15668040696168
// MI455X (gfx1250) — compile-verified
//
#include <hip/hip_runtime.h>

// Problem constants (from reference): u[B=4, I=32, N=32, J=16, H=32, W=32] f32
#define BB 4
#define II 32
#define NN 32
#define JJ 16
#define HH 32
#define WW 32
#define W2 2                      // w-pair per workgroup (8B contiguous)
#define NWQ (WW / W2)             // 16
#define LSTRIDE (HH * WW)         // 1024 floats between consecutive (i,n,j) rows
#define NROWS (II * NN)           // 1024 (i,n) rows
#define NL (NROWS * JJ)           // 16384 8-byte chunks per workgroup

// Padded LDS strides (floats), chosen for conflict-free b64 access:
//  ROWP=34: gcd(34,64)=2 -> {34*n mod 64} distinct for n=0..31 (phase 2/4)
//  CROWP=66: {66*i mod 64}={2i} distinct for i=0..31 (phase 3 softmax)
#define ROWP (JJ * W2 + 2)        // 34 floats per (i,n) row of u-tile
#define VROWP (JJ * W2 + 2)       // 34 floats per n row of v / s
#define CROWP (NN * W2 + 2)       // 66 floats per i row of c / softmax(c)

#define U_OFF 0
#define VS_OFF (U_OFF + NROWS * ROWP)    // 34816
#define CB_OFF (VS_OFF + NN * VROWP)     // 35904
#define SC_OFF (CB_OFF + II * CROWP)     // 38016
#define LDS_FLOATS (SC_OFF + NN * W2)    // 38080 floats = 152320 B (~149 KB)
                                         // => 2 workgroups per 320 KB WGP

#define THREADS 512
#define EPSF 1e-20f

// Async global->LDS copy engine (arg pattern confirmed for b128 by clang
// diagnostic: (vNi addrspace(1)* src, vNi addrspace(3)* dst, imm off, imm cpol))
typedef int v2i __attribute__((vector_size(8)));
typedef __attribute__((address_space(1))) v2i* g_v2i_p;
typedef __attribute__((address_space(3))) v2i* l_v2i_p;

#if __has_builtin(__builtin_amdgcn_global_load_async_to_lds_b64)
#define HAVE_ASYNC_LDS 1
#else
#define HAVE_ASYNC_LDS 0
#warning "gfx1250: __builtin_amdgcn_global_load_async_to_lds_b64 NOT available; using sync copy fallback"
#endif

#if __has_builtin(__builtin_amdgcn_s_wait_asynccnt)
#define WAIT_ASYNC0() __builtin_amdgcn_s_wait_asynccnt(0)
#else
#define WAIT_ASYNC0() asm volatile("s_wait_asynccnt 0x0" ::: "memory")
#endif

__global__ __launch_bounds__(THREADS, 2)
void caps_route_kernel(const float* __restrict__ u,
                       const float* __restrict__ bias,   // [I*N], broadcast over j,h,w
                       float* __restrict__ out) {        // [B,N,J,H,W]
  extern __shared__ float sm[];
  const int t = threadIdx.x;
  const int blk = blockIdx.x;          // ((b*HH + h)*NWQ + wq), wq fastest for L2 sharing
  const int wq = blk & (NWQ - 1);
  const int h = (blk >> 4) & (HH - 1);
  const int b = blk >> 9;

  // element index of u[b, 0, 0, 0, h, wq*2]
  const long long gbase =
      (long long)b * (II * NN * JJ * HH * WW) + (long long)h * WW + wq * W2;
  const float* gp = u + gbase;

  // ---------------- Phase 0: stage 128 KB tile into LDS (async copy) -------
  // chunk L = (i*NN + n)*JJ + j ; 8 contiguous bytes (2 w's) per chunk
#pragma unroll
  for (int k = 0; k < NL / THREADS; ++k) {   // 32 chunks per thread
    const int L = t + k * THREADS;
    const int row = L >> 4;                  // i*NN + n
    const int j = L & 15;
    const float* src = gp + (long long)L * LSTRIDE;
    float* dst = &sm[U_OFF + row * ROWP + j * W2];
#if HAVE_ASYNC_LDS
    __builtin_amdgcn_global_load_async_to_lds_b64((g_v2i_p)src, (l_v2i_p)dst,
                                                  /*offset=*/0, /*cpol=*/0);
#else
    *(float2*)dst = *(const float2*)src;
#endif
  }
#if HAVE_ASYNC_LDS
  WAIT_ASYNC0();
#endif
  __syncthreads();

  // ---------------- Phase 1: v[n,j,0:2] = sum_i u[i,n,j,0:2] ----------------
  {                                           // 512 threads: (n, j), float2 over w
    const int j = t & 15;
    const int n = t >> 4;
    float2 acc = make_float2(0.f, 0.f);
#pragma unroll
    for (int i = 0; i < II; ++i) {
      const float2 a = *(const float2*)&sm[U_OFF + (i * NN + n) * ROWP + j * W2];
      acc.x += a.x; acc.y += a.y;
    }
    *(float2*)&sm[VS_OFF + n * VROWP + j * W2] = acc;
  }
  __syncthreads();

  // ---------------- Phase 2: c[i,n,0:2] = sum_j u[i,n,j,0:2]*v[n,j,0:2] -----
  {
    const int n = t & 31;
    const int i0 = t >> 5;                    // 0..15; two i per thread
#pragma unroll
    for (int it = 0; it < 2; ++it) {
      const int i = i0 + 16 * it;
      const float2* urow = (const float2*)&sm[U_OFF + (i * NN + n) * ROWP];
      const float2* vrow = (const float2*)&sm[VS_OFF + n * VROWP];
      float2 acc = make_float2(0.f, 0.f);
#pragma unroll
      for (int j = 0; j < JJ; ++j) {
        const float2 a = urow[j];
        const float2 vv = vrow[j];
        acc.x += a.x * vv.x;
        acc.y += a.y * vv.y;
      }
      *(float2*)&sm[CB_OFF + i * CROWP + n * W2] = acc;
    }
  }
  __syncthreads();

  // ---------------- Phase 3: softmax over n of c/4, plus bias ---------------
  if (t < II) {                               // 32 threads: one i each, float2 over w
    const int i = t;
    float2* crow = (float2*)&sm[CB_OFF + i * CROWP];
    float2 m = make_float2(-3.4e38f, -3.4e38f);
#pragma unroll
    for (int n = 0; n < NN; ++n) {
      const float2 l = crow[n];
      m.x = fmaxf(m.x, l.x * 0.25f);
      m.y = fmaxf(m.y, l.y * 0.25f);
    }
    float2 d = make_float2(0.f, 0.f);
#pragma unroll
    for (int n = 0; n < NN; ++n) {
      const float2 l = crow[n];
      d.x += expf(l.x * 0.25f - m.x);
      d.y += expf(l.y * 0.25f - m.y);
    }
    const float2 inv = make_float2(1.f / d.x, 1.f / d.y);
#pragma unroll
    for (int n = 0; n < NN; ++n) {
      const float2 l = crow[n];
      const float bb = bias[i * NN + n];      // broadcast over w
      float2 e;
      e.x = expf(l.x * 0.25f - m.x) * inv.x + bb;
      e.y = expf(l.y * 0.25f - m.y) * inv.y + bb;
      crow[n] = e;                            // overwrite c with routing coeff
    }
  }
  __syncthreads();

  // ---------------- Phase 4: s[n,j,0:2] = sum_i u[i,n,j,0:2]*c[i,n,0:2] -----
  {                                           // 512 threads: (n, j)
    const int j = t & 15;
    const int n = t >> 4;
    float2 acc = make_float2(0.f, 0.f);
#pragma unroll
    for (int i = 0; i < II; ++i) {
      const float2 a = *(const float2*)&sm[U_OFF + (i * NN + n) * ROWP + j * W2];
      const float2 c = *(const float2*)&sm[CB_OFF + i * CROWP + n * W2];
      acc.x += a.x * c.x;
      acc.y += a.y * c.y;
    }
    *(float2*)&sm[VS_OFF + n * VROWP + j * W2] = acc;  // s reuses v's buffer
  }
  __syncthreads();

  // ---------------- Phase 5: squash scale per (n, 0:2) over j ---------------
  if (t < NN) {                               // 32 threads: one n each
    const int n = t;
    float2 ss = make_float2(0.f, 0.f);
#pragma unroll
    for (int j = 0; j < JJ; ++j) {
      const float2 x = *(const float2*)&sm[VS_OFF + n * VROWP + j * W2];
      ss.x += x.x * x.x;
      ss.y += x.y * x.y;
    }
    float2 sc;
    const float nx = sqrtf(ss.x), ny = sqrtf(ss.y);
    sc.x = (1.0f - 1.0f / (expf(nx) + EPSF)) / (nx + EPSF);
    sc.y = (1.0f - 1.0f / (expf(ny) + EPSF)) / (ny + EPSF);
    *(float2*)&sm[SC_OFF + n * W2] = sc;
  }
  __syncthreads();

  // ---------------- Phase 6: scaled write-out [B,N,J,H,W] -------------------
  {                                           // 512 threads: (n, j), float2 over w
    const int n = t >> 4;
    const int j = t & 15;
    const float2 sv = *(const float2*)&sm[VS_OFF + n * VROWP + j * W2];
    const float2 sc = *(const float2*)&sm[SC_OFF + n * W2];
    float2 o;
    o.x = sv.x * sc.x;
    o.y = sv.y * sc.y;
    const long long oidx = (long long)b * (NN * JJ * HH * WW) +
                           (long long)(n * JJ + j) * (HH * WW) +
                           (long long)h * WW + wq * W2;
    *(float2*)(out + oidx) = o;
  }
}

extern "C" void kernel_launch(void* const* d_in, const int* in_sizes, int n_in,
                              void* d_out, int out_size, void* d_ws, size_t ws_size,
                              hipStream_t stream) {
  const float* u = (const float*)d_in[0];     // [4,32,32,16,32,32] f32
  const float* bias = (const float*)d_in[1];  // [1,32,32,1,1,1] f32
  float* out = (float*)d_out;                 // [4,32,16,32,32] f32

  const dim3 grid(BB * HH * NWQ);             // 2048 workgroups
  const dim3 block(THREADS);                  // 512 threads = 16 wave32
  const size_t lds_bytes = (size_t)LDS_FLOATS * sizeof(float);  // ~149 KB -> 2 WG/WGP

  caps_route_kernel<<<grid, block, lds_bytes, stream>>>(u, bias, out);
}